// KConditionalRandomField_18442589570254
// MI455X (gfx1250) — compile-verified
//
#include <hip/hip_runtime.h>
#include <cstdint>

// Problem constants (from reference): B=256, T=512, L=128
#define CB 256
#define CT 512
#define CL 128

typedef __attribute__((ext_vector_type(4))) unsigned int u32x4;
typedef __attribute__((ext_vector_type(8))) int          i32x8;
typedef __attribute__((ext_vector_type(4))) int          i32x4;

// ---------------------------------------------------------------------------
// Kernel 1: Viterbi forward. One workgroup per batch, one thread per "to"
// label j. Transitions (64KB) DMA'd into LDS once via the Tensor Data Mover,
// state ping-ponged in LDS (1 barrier/step). Backpointers stored as uint8
// overlaid into this batch's region of d_out; final argmax tag stored after.
// ---------------------------------------------------------------------------
__global__ __launch_bounds__(CL) void crf_viterbi_fwd(
    const float* __restrict__ pots,       // [B,T,L]
    const float* __restrict__ trans,      // [L,L]
    const unsigned char* __restrict__ mask, // [B,T] (bool bytes)
    float* __restrict__ out)              // [B,T,L] (used as byte scratch here)
{
    extern __shared__ float smem[];
    float* trans_s = smem;                 // 128*128 floats = 64KB
    float* state_s = smem + CL * CL;       // 2*128 floats (ping-pong)

    const int b = blockIdx.x;
    const int j = threadIdx.x;             // 0..127

    // --- TDM: DMA the 128x128 f32 transitions tile global -> LDS (wave 0) ---
    if (threadIdx.x < 32) {                // wave-uniform branch (wave32)
        uint32_t lds_off = (uint32_t)(uintptr_t)(void*)trans_s; // low 32 bits = LDS byte offset
        uint64_t ga = (uint64_t)(uintptr_t)(const void*)trans;

        u32x4 g0;
        g0[0] = 1u;                                   // count=1 (valid), user mode, no gather
        g0[1] = lds_off;                              // lds_addr
        g0[2] = (uint32_t)(ga & 0xFFFFFFFFu);         // global_addr[31:0]
        g0[3] = (uint32_t)((ga >> 32) & 0x1FFFFFFu)   // global_addr[56:32]
              | (2u << 30);                           // type = 2 ("image")

        i32x8 g1;
        g1[0] = (int)(2u << 16);        // workgroup_mask=0, data_size=2 (4 bytes)
        g1[1] = (int)(128u << 16);      // tensor_dim0 = 128 (low 16 in bits[31:16])
        g1[2] = (int)(128u << 16);      // tensor_dim0 hi=0 | tensor_dim1 = 128
        g1[3] = (int)(128u << 16);      // tensor_dim1 hi=0 | tile_dim0 = 128
        g1[4] = 128;                    // tile_dim1 = 128, tile_dim2 = 0
        g1[5] = 128;                    // tensor_dim0_stride = 128 (elements)
        g1[6] = (int)(16384u << 16);    // dim0_stride hi=0 | tensor_dim1_stride lo = 16384
        g1[7] = 0;

        i32x4 gz4 = {0, 0, 0, 0};              // groups 2/3 unused (2D tensor)
        i32x8 gz8 = {0, 0, 0, 0, 0, 0, 0, 0};  // extra group (clang-23 6-arg form)
        __builtin_amdgcn_tensor_load_to_lds(g0, g1, gz4, gz4, gz8, 0);
        __builtin_amdgcn_s_wait_tensorcnt(0);
    }

    // init state with pots[b,0,:]
    const size_t base = (size_t)b * CT * CL;
    state_s[j] = pots[base + j];
    __syncthreads();                       // TDM data + init state visible

    unsigned char* bp_base = (unsigned char*)out + base * sizeof(float);

    int p = 0;
    float pot_next = pots[base + CL + j];  // prefetch t=1
    for (int t = 1; t < CT; ++t) {
        const float pot = pot_next;
        if (t + 1 < CT) pot_next = pots[base + (size_t)(t + 1) * CL + j];
        const unsigned char m = mask[(size_t)b * CT + t];

        const float* st = state_s + p * CL;
        float best = -__builtin_inff();
        int   bi   = 0;
#pragma unroll 8
        for (int i = 0; i < CL; ++i) {
            // st[i]: LDS broadcast; trans_s[i*CL+j]: conflict-free (bank = j mod 64)
            float v = st[i] + trans_s[i * CL + j];
            if (v > best) { best = v; bi = i; }   // strict '>' keeps first argmax
        }
        float ns  = m ? (best + pot) : st[j];
        int   bpv = m ? bi : j;                   // identity backpointer when masked

        state_s[(1 - p) * CL + j] = ns;
        bp_base[(size_t)(t - 1) * CL + j] = (unsigned char)bpv;
        p = 1 - p;
        __syncthreads();
    }

    // argmax of final state (one-time 128-iteration scan; negligible)
    if (j == 0) {
        const float* st = state_s + p * CL;
        float best = st[0]; int bi = 0;
        for (int i = 1; i < CL; ++i)
            if (st[i] > best) { best = st[i]; bi = i; }
        *(int*)(bp_base + (size_t)(CT - 1) * CL) = bi;  // byte offset 65408 (4B aligned)
    }
}

// ---------------------------------------------------------------------------
// Kernel 2: backtrace + one-hot. Backpointer rows are staged into LDS in
// 4KB chunks (row addresses don't depend on the tag), so the dependent
// pointer-chase runs at LDS latency. Then all threads emit the one-hot,
// fully overwriting the scratch bytes.
// ---------------------------------------------------------------------------
__global__ __launch_bounds__(CL) void crf_backtrace_onehot(float* __restrict__ out)
{
    __shared__ unsigned char rows[32 * CL];   // 4KB chunk of backpointer rows
    __shared__ short tags[CT];                // decoded tag sequence

    const int b   = blockIdx.x;
    const int tid = threadIdx.x;              // 0..127
    const size_t base = (size_t)b * CT * CL;
    const unsigned char* bp_base = (const unsigned char*)out + base * sizeof(float);

    int tag = 0;
    if (tid == 0) tag = *(const int*)(bp_base + (size_t)(CT - 1) * CL);

    // k runs 510..0 (bps index); chunk c covers k in [32c, 32c+31]
    for (int c = 15; c >= 0; --c) {
        const int k0 = c * 32;
        const int nrows = (CT - 1 - k0) < 32 ? (CT - 1 - k0) : 32;  // last chunk: 31
        for (int r = 0; r < nrows; ++r)
            rows[r * CL + tid] = bp_base[(size_t)(k0 + r) * CL + tid];
        __syncthreads();
        if (tid == 0) {
            for (int k = k0 + nrows - 1; k >= k0; --k) {
                tags[k + 1] = (short)tag;          // y_{k+1} = current tag
                tag = rows[(k - k0) * CL + tag];   // follow backpointer
            }
        }
        __syncthreads();                           // rows reused next chunk
    }
    if (tid == 0) tags[0] = (short)tag;
    __syncthreads();

    // one-hot expansion: adjacent tids -> adjacent addresses (coalesced)
    for (int t = 0; t < CT; ++t)
        out[base + (size_t)t * CL + tid] = (tags[t] == tid) ? 1.0f : 0.0f;
}

// ---------------------------------------------------------------------------
extern "C" void kernel_launch(void* const* d_in, const int* in_sizes, int n_in,
                              void* d_out, int out_size, void* d_ws, size_t ws_size,
                              hipStream_t stream) {
    const float* pots          = (const float*)d_in[0];          // [B,T,L] f32
    const float* trans         = (const float*)d_in[1];          // [L,L]   f32
    const unsigned char* mask  = (const unsigned char*)d_in[2];  // [B,T]   bool
    float* out                 = (float*)d_out;                  // [B,T,L] f32

    const size_t sh = (size_t)(CL * CL + 2 * CL) * sizeof(float); // 65KB dynamic LDS
    crf_viterbi_fwd<<<CB, CL, sh, stream>>>(pots, trans, mask, out);
    crf_backtrace_onehot<<<CB, CL, 0, stream>>>(out);
}